// SA_attention_30794915512878
// MI455X (gfx1250) — compile-verified
//
#include <hip/hip_runtime.h>

// ---- problem constants -------------------------------------------------
#define NB     4
#define CC     256
#define HEADS  4
#define DH     64
#define HW     4096
#define QKVF   768            // 3*C
#define SCALE  0.0625f        // C^-0.5 = 1/16

#define KSTR   68             // K-tile LDS row stride (16B aligned, bank-spread)
#define VSTR   17             // V^T tile LDS row stride

typedef float v8f __attribute__((ext_vector_type(8)));
typedef float v2f __attribute__((ext_vector_type(2)));

// Native fp32 matrix op: D(16x16) = A(16x4) * B(4x16) + C
__device__ __forceinline__ v8f wmma_f32(v2f a, v2f b, v8f c) {
  return __builtin_amdgcn_wmma_f32_16x16x4_f32(
      /*neg_a=*/false, a, /*neg_b=*/false, b,
      /*c_mod=*/(short)0, c, /*reuse_a=*/false, /*reuse_b=*/false);
}

// CDNA5 async global->LDS copies (tracked by ASYNCcnt, no VGPR round-trip)
__device__ __forceinline__ void async_b128(unsigned lds_off, const float* g) {
  asm volatile("global_load_async_to_lds_b128 %0, %1, off"
               :: "v"(lds_off), "v"(g) : "memory");
}
__device__ __forceinline__ void async_b32(unsigned lds_off, const float* g) {
  asm volatile("global_load_async_to_lds_b32 %0, %1, off"
               :: "v"(lds_off), "v"(g) : "memory");
}
__device__ __forceinline__ void wait_async0() {
  asm volatile("s_wait_asynccnt 0x0" ::: "memory");
}

// ======================================================================
// Kernel 1: qkv = x^T @ W + b ; split into Q (mask*scale folded), K, V
// laid out as (n*heads, hw, 64) row-major in workspace.
// One wave = one 16x16 output tile, K=256 -> 64 WMMAs.
// ======================================================================
__global__ void __launch_bounds__(256)
qkv_proj_kernel(const float* __restrict__ x, const float* __restrict__ mask,
                const float* __restrict__ W, const float* __restrict__ bias,
                float* __restrict__ Qd, float* __restrict__ Kd,
                float* __restrict__ Vd)
{
  const int lane = threadIdx.x & 31;
  const int wv   = threadIdx.x >> 5;
  const int gid  = blockIdx.x * 8 + wv;

  const int TPN = (HW / 16) * (QKVF / 16);      // tiles per batch = 12288
  const int n   = gid / TPN;
  const int r   = gid % TPN;
  const int p0  = (r / (QKVF / 16)) * 16;       // token tile
  const int f0  = (r % (QKVF / 16)) * 16;       // output-feature tile
  const int row = lane & 15;                    // M-row / N-col (per operand)
  const int half = lane >> 4;

  const float* xb = x + (size_t)n * CC * HW;

  v8f acc = {};
#pragma unroll 4
  for (int k0 = 0; k0 < CC; k0 += 4) {
    const int kb = k0 + half * 2;
    v2f a, b;
    // A[p][k] = x[k][p]  (x is channel-major -> tokens contiguous, coalesced)
    a.x = xb[(size_t)kb * HW + p0 + row];
    a.y = xb[(size_t)(kb + 1) * HW + p0 + row];
    // B[k][f] = W[k][f]
    b.x = W[(size_t)kb * QKVF + f0 + row];
    b.y = W[(size_t)(kb + 1) * QKVF + f0 + row];
    acc = wmma_f32(a, b, acc);
  }

  // which tensor / head this 16-wide feature tile belongs to (tile-uniform)
  const int which = f0 >> 8;        // 0=Q 1=K 2=V
  const int fr    = f0 & 255;
  const int head  = fr >> 6;
  const int d0    = fr & 63;
  float* dst = (which == 0) ? Qd : (which == 1) ? Kd : Vd;
  const size_t base = (size_t)(n * HEADS + head) * HW * DH;
  const float  bq   = bias[f0 + row];

#pragma unroll
  for (int i = 0; i < 8; ++i) {
    const int p = p0 + i + 8 * half;            // C-layout: M = i (+8 hi half)
    float v = acc[i] + bq;
    if (which == 0) v *= mask[n * HW + p] * SCALE;   // fold mask & 1/sqrt(C)
    dst[base + (size_t)p * DH + d0 + row] = v;
  }
}

// ======================================================================
// Kernel 2: flash attention. Grid (HW/128, NB*HEADS), 256 threads = 8 waves.
// Each wave owns 16 queries; loops over 16-key blocks with online softmax.
// K/V tiles are double-buffered in LDS via async global->LDS copies, so
// the next block's loads overlap the current block's WMMA/softmax work.
// ======================================================================
__global__ void __launch_bounds__(256)
attn_kernel(const float* __restrict__ Qd, const float* __restrict__ Kd,
            const float* __restrict__ Vd, float* __restrict__ out)
{
  __shared__ __attribute__((aligned(16))) float Kt[2][16 * KSTR]; // K row-major
  __shared__ __attribute__((aligned(16))) float Vt[2][64 * VSTR]; // V^T (d x key)
  __shared__ float Pl[8][16 * VSTR];                              // per-wave P^T

  const int tid  = threadIdx.x;
  const int lane = tid & 31;
  const int wv   = tid >> 5;
  const int col  = lane & 15;          // query column / key row (lane%16)
  const int half = lane >> 4;
  const int bh   = blockIdx.y;
  const int n    = bh >> 2;
  const int head = bh & 3;
  const int q0   = blockIdx.x * 128 + wv * 16;

  const float* Qp = Qd + (size_t)bh * HW * DH;
  const float* Kp = Kd + (size_t)bh * HW * DH;
  const float* Vp = Vd + (size_t)bh * HW * DH;

  // async stage of one 16-key block into buffer `buf`
  const int skey = tid >> 4, sch4 = (tid & 15) * 4;   // K: one b128 per thread
  auto stage = [&](int buf, int m0) {
    async_b128((unsigned)(uintptr_t)&Kt[buf][skey * KSTR + sch4],
               Kp + (size_t)(m0 + skey) * DH + sch4);
#pragma unroll
    for (int i = 0; i < 4; ++i) {                     // V: 4 scattered b32
      const int idx = tid + i * 256;
      const int key = idx >> 6, ch = idx & 63;
      async_b32((unsigned)(uintptr_t)&Vt[buf][ch * VSTR + key],
                Vp + (size_t)(m0 + key) * DH + ch);
    }
  };

  // B-operands of S^T = K_blk @ Q^T: {Q[q][kb], Q[q][kb+1]}, kb = 4c + 2*half
  v2f qb[16];
  {
    const float* qrow = Qp + (size_t)(q0 + col) * DH + half * 2;
#pragma unroll
    for (int c = 0; c < 16; ++c) {
      qb[c].x = qrow[c * 4 + 0];
      qb[c].y = qrow[c * 4 + 1];
    }
  }

  float mrun = -3.40282e38f, lrun = 0.f;
  v8f o0 = {}, o1 = {}, o2 = {}, o3 = {};   // O^T: d-rows {0..15,...,48..63}

  // preload block 0
  stage(0, 0);
  wait_async0();
  __syncthreads();

  for (int m0 = 0, it = 0; m0 < HW; m0 += 16, ++it) {
    const int cur = it & 1;
    if (m0 + 16 < HW) stage(cur ^ 1, m0 + 16);   // overlap next block's loads

    const float* kt = Kt[cur];
    const float* vt = Vt[cur];

    // S^T (16 keys x 16 queries) = K_blk(16x64) @ Q^T(64x16)
    v8f s = {};
#pragma unroll
    for (int c = 0; c < 16; ++c) {
      const int kb = c * 4 + half * 2;
      v2f a;
      a.x = kt[col * KSTR + kb];
      a.y = kt[col * KSTR + kb + 1];
      s = wmma_f32(a, qb[c], s);
    }

    // ---- online softmax along keys (per query == per lane) ----
    float bm = s[0];
#pragma unroll
    for (int i = 1; i < 8; ++i) bm = fmaxf(bm, s[i]);
    bm = fmaxf(bm, __shfl_xor(bm, 16, 32));      // combine key halves
    const float nm    = fmaxf(mrun, bm);
    const float alpha = __expf(mrun - nm);
    float ps = 0.f;
    v8f p;
#pragma unroll
    for (int i = 0; i < 8; ++i) { p[i] = __expf(s[i] - nm); ps += p[i]; }
    ps += __shfl_xor(ps, 16, 32);
    lrun = lrun * alpha + ps;
    mrun = nm;
#pragma unroll
    for (int i = 0; i < 8; ++i) {
      o0[i] *= alpha; o1[i] *= alpha; o2[i] *= alpha; o3[i] *= alpha;
    }

    // P^T (C-layout) -> per-wave LDS row-major 16x16 (stride 17)
    float* pl = Pl[wv];
#pragma unroll
    for (int i = 0; i < 8; ++i) pl[(i + 8 * half) * VSTR + col] = p[i];
    asm volatile("s_wait_dscnt 0" ::: "memory");

    // O^T(64x16) += V^T(64x16keys) @ P^T(16keys x 16q), K=16 -> 4 chunks
#pragma unroll
    for (int c = 0; c < 4; ++c) {
      const int kb = c * 4 + half * 2;
      v2f b;
      b.x = pl[kb * VSTR + col];
      b.y = pl[(kb + 1) * VSTR + col];
      v2f a;
      a.x = vt[col * VSTR + kb];        a.y = vt[col * VSTR + kb + 1];
      o0 = wmma_f32(a, b, o0);
      a.x = vt[(16 + col) * VSTR + kb]; a.y = vt[(16 + col) * VSTR + kb + 1];
      o1 = wmma_f32(a, b, o1);
      a.x = vt[(32 + col) * VSTR + kb]; a.y = vt[(32 + col) * VSTR + kb + 1];
      o2 = wmma_f32(a, b, o2);
      a.x = vt[(48 + col) * VSTR + kb]; a.y = vt[(48 + col) * VSTR + kb + 1];
      o3 = wmma_f32(a, b, o3);
    }

    wait_async0();      // next buffer's async copies have landed
    __syncthreads();    // everyone done reading `cur` and done waiting
  }

  // normalize and write out[n][head*64+ch][q0+col] (16-lane contiguous)
  const float rl = 1.0f / lrun;
  float* ob = out + ((size_t)n * CC + head * DH) * HW + q0 + col;
#pragma unroll
  for (int i = 0; i < 8; ++i) {
    const int chb = i + 8 * half;
    ob[(size_t)chb * HW]        = o0[i] * rl;
    ob[(size_t)(chb + 16) * HW] = o1[i] * rl;
    ob[(size_t)(chb + 32) * HW] = o2[i] * rl;
    ob[(size_t)(chb + 48) * HW] = o3[i] * rl;
  }
}

// ======================================================================
extern "C" void kernel_launch(void* const* d_in, const int* in_sizes, int n_in,
                              void* d_out, int out_size, void* d_ws,
                              size_t ws_size, hipStream_t stream) {
  const float* x    = (const float*)d_in[0];
  const float* mask = (const float*)d_in[1];
  const float* Wq   = (const float*)d_in[2];
  const float* bq   = (const float*)d_in[3];
  float* out = (float*)d_out;

  const size_t per = (size_t)NB * HEADS * HW * DH;   // 4,194,304 floats
  float* Qd = (float*)d_ws;
  float* Kd = Qd + per;
  float* Vd = Kd + per;

  // kernel 1: 49152 16x16 tiles, 8 waves per block -> 6144 blocks
  qkv_proj_kernel<<<dim3(6144), dim3(256), 0, stream>>>(x, mask, Wq, bq,
                                                        Qd, Kd, Vd);
  // kernel 2: 32 query-blocks x 16 batch-heads
  attn_kernel<<<dim3(HW / 128, NB * HEADS), dim3(256), 0, stream>>>(Qd, Kd,
                                                                    Vd, out);
}